// Attn_61366492725428
// MI455X (gfx1250) — compile-verified
//
#include <hip/hip_runtime.h>
#include <stdint.h>

// Problem constants (from reference)
constexpr int CB    = 32;    // batch
constexpr int CTOK  = 2048;  // T
constexpr int CHE   = 1024;  // encoder hidden
// Tiling
constexpr int CNC   = 16;            // T-chunks per batch -> 512 workgroups
constexpr int CCT   = CTOK / CNC;    // 128 timesteps per chunk
constexpr int CTT   = 16;            // tile rows (matches WMMA M=16)
constexpr int CNTIL = CCT / CTT;     // 8 tiles per chunk
constexpr int CPITCH = 1028;         // padded floats per LDS row (bank spread)
constexpr int CTHR  = 256;           // 8 wave32s
constexpr int CWAVE = 8;
constexpr int CKS   = CHE / CWAVE;   // 128-wide K-slice per wave

typedef float v2f __attribute__((ext_vector_type(2)));
typedef float v8f __attribute__((ext_vector_type(8)));

// Dynamic LDS layout (floats): [2 tile buffers][We][chunk mask][wave partials][s final]
constexpr int SM_TILES  = 2 * CTT * CPITCH;
constexpr int SM_FLOATS = SM_TILES + CHE + CCT + CWAVE * 16 + 16;
constexpr size_t SM_BYTES = (size_t)SM_FLOATS * 4;

__global__ __launch_bounds__(CTHR) void attn_pass1(
    const float* __restrict__ enc, const float* __restrict__ mask,
    const float* __restrict__ W, float* __restrict__ ws)
{
    extern __shared__ float smem[];
    float* tiles = smem;                 // [2][CTT][CPITCH]
    float* Wel   = smem + SM_TILES;      // [CHE]
    float* mloc  = Wel + CHE;            // [CCT]
    float* sp    = mloc + CCT;           // [CWAVE][16] per-wave energy partials
    float* sfin  = sp + CWAVE * 16;      // [16]

    const int tid = threadIdx.x;
    const int b   = blockIdx.y;
    const int c   = blockIdx.x;
    const int t0c = c * CCT;
    const float* encb = enc + (size_t)b * CTOK * CHE;

    // Stage We (= W[1024:2048]) and this chunk's mask row into LDS.
    #pragma unroll
    for (int i = 0; i < CHE / CTHR; ++i)
        Wel[tid + i * CTHR] = W[1024 + tid + i * CTHR];
    if (tid < CCT) mloc[tid] = mask[(size_t)b * CTOK + t0c + tid];

    const uint32_t tile_lds0 = (uint32_t)(uintptr_t)tiles;  // low 32 bits = LDS byte offset
    const unsigned long long gb = (unsigned long long)(uintptr_t)encb;

    // Async global -> LDS copy of one 16x1024 f32 tile (64 KB), 16 b128 per thread.
    auto issue_tile = [&](int tileIdx, int bufIdx) {
        const uint32_t ldsbase = tile_lds0 + (uint32_t)(bufIdx * CTT * CPITCH * 4);
        #pragma unroll
        for (int k = 0; k < 16; ++k) {
            int f   = k * CTHR + tid;   // float4 index within tile
            int row = f >> 8;           // 256 float4 per row
            int c4  = f & 255;
            uint32_t lds = ldsbase + (uint32_t)(row * CPITCH * 4 + c4 * 16);
            uint32_t go  = (uint32_t)((((t0c + tileIdx * CTT + row) * CHE) + c4 * 4) * 4);
            asm volatile("global_load_async_to_lds_b128 %0, %1, %2"
                         :: "v"(lds), "v"(go), "s"(gb) : "memory");
        }
    };

    issue_tile(0, 0);

    const int lane  = tid & 31;
    const int w     = tid >> 5;
    const int kbase = w * CKS;
    const int arow  = lane & 15;          // A-matrix row this lane holds
    const int khalf = (lane >> 4) << 1;   // lanes 16-31 hold K+2,K+3

    float run_m = -3.0e38f, run_l = 0.f;
    float acc0 = 0.f, acc1 = 0.f, acc2 = 0.f, acc3 = 0.f;  // context for h = 4*tid..+3

    for (int it = 0; it < CNTIL; ++it) {
        if (it + 1 < CNTIL) {
            issue_tile(it + 1, (it + 1) & 1);
            asm volatile("s_wait_asynccnt 16" ::: "memory"); // tile `it` (oldest 16 ops) done
        } else {
            asm volatile("s_wait_asynccnt 0" ::: "memory");
        }
        __syncthreads();

        const float* tile = tiles + (it & 1) * CTT * CPITCH;

        // ---- energies s[t] = enc[t,:]·We : per-wave K-slice via WMMA f32 16x16x4 ----
#if __has_builtin(__builtin_amdgcn_wmma_f32_16x16x4_f32)
        {
            const float* trow = tile + arow * CPITCH;
            v8f d = {0.f, 0.f, 0.f, 0.f, 0.f, 0.f, 0.f, 0.f};
            #pragma unroll
            for (int kk = 0; kk < CKS / 4; ++kk) {
                const int k = kbase + kk * 4 + khalf;
                v2f a; a.x = trow[k]; a.y = trow[k + 1];     // A: 16x4 f32 layout
                v2f bb; bb.x = Wel[k]; bb.y = Wel[k + 1];    // B: We replicated -> all N cols equal
                d = __builtin_amdgcn_wmma_f32_16x16x4_f32(
                        false, a, false, bb, (short)0, d, false, false);
            }
            // D layout: lane0 holds rows 0..7 (N=0), lane16 holds rows 8..15 (N=0)
            if (lane == 0) {
                #pragma unroll
                for (int r = 0; r < 8; ++r) sp[w * 16 + r] = d[r];
            } else if (lane == 16) {
                #pragma unroll
                for (int r = 0; r < 8; ++r) sp[w * 16 + 8 + r] = d[r];
            }
        }
#else
        {   // VALU fallback: lane covers 4 cols of the wave's K-slice
            #pragma unroll
            for (int t = 0; t < CTT; ++t) {
                const float* tr = tile + t * CPITCH + kbase + lane * 4;
                const float* wr = Wel + kbase + lane * 4;
                float p = tr[0]*wr[0] + tr[1]*wr[1] + tr[2]*wr[2] + tr[3]*wr[3];
                #pragma unroll
                for (int off = 16; off > 0; off >>= 1) p += __shfl_xor(p, off, 32);
                if (lane == 0) sp[w * 16 + t] = p;
            }
        }
#endif
        __syncthreads();
        if (tid < 16) {
            float s = 0.f;
            #pragma unroll
            for (int ww = 0; ww < CWAVE; ++ww) s += sp[ww * 16 + tid];
            sfin[tid] = s;
        }
        __syncthreads();

        // ---- online masked softmax update (redundant, identical on all threads) ----
        float sv[CTT], mv[CTT];
        float tmax = -3.0e38f;
        #pragma unroll
        for (int t = 0; t < CTT; ++t) {
            sv[t] = sfin[t];
            mv[t] = mloc[it * CTT + t];
            if (mv[t] > 0.f) tmax = fmaxf(tmax, sv[t]);
        }
        if (tmax > -1.0e37f) {                 // tile has at least one unmasked step
            float newm  = fmaxf(run_m, tmax);
            float scale = __expf(run_m - newm); // 0 on first update
            run_l *= scale;
            acc0 *= scale; acc1 *= scale; acc2 *= scale; acc3 *= scale;
            float wt[CTT];
            #pragma unroll
            for (int t = 0; t < CTT; ++t) {
                wt[t] = (mv[t] > 0.f) ? __expf(sv[t] - newm) : 0.f;
                run_l += wt[t];
            }
            run_m = newm;
            const float* tcol = tile + tid * 4;
            #pragma unroll
            for (int t = 0; t < CTT; ++t) {
                const float4 e = *(const float4*)(tcol + t * CPITCH);
                acc0 += wt[t] * e.x; acc1 += wt[t] * e.y;
                acc2 += wt[t] * e.z; acc3 += wt[t] * e.w;
            }
        }
        __syncthreads();  // reads of this buffer done before it is async-overwritten
    }

    // Workspace: header[(b*NC+c)*2] = {m,l}; acc at ws+1024, 1024 f32 per chunk.
    if (tid == 0) {
        ws[(b * CNC + c) * 2]     = run_m;
        ws[(b * CNC + c) * 2 + 1] = run_l;
    }
    float* accw = ws + 1024 + ((size_t)(b * CNC + c)) * 1024;
    float4 av = make_float4(acc0, acc1, acc2, acc3);
    *(float4*)&accw[tid * 4] = av;
}

__global__ __launch_bounds__(CTHR) void attn_pass2(
    const float* __restrict__ ws, float* __restrict__ out)
{
    __shared__ float sm[CNC], sl[CNC];
    const int b = blockIdx.x, tid = threadIdx.x;
    if (tid < CNC) {
        sm[tid] = ws[(b * CNC + tid) * 2];
        sl[tid] = ws[(b * CNC + tid) * 2 + 1];
    }
    __syncthreads();
    float M = -3.0e38f;
    #pragma unroll
    for (int c = 0; c < CNC; ++c) M = fmaxf(M, sm[c]);
    float L = 0.f;
    #pragma unroll
    for (int c = 0; c < CNC; ++c) L += sl[c] * __expf(sm[c] - M);
    const float* accw = ws + 1024 + (size_t)b * CNC * 1024;
    float4 acc = make_float4(0.f, 0.f, 0.f, 0.f);
    #pragma unroll
    for (int c = 0; c < CNC; ++c) {
        float wc = __expf(sm[c] - M);
        float4 a = *(const float4*)&accw[(size_t)c * 1024 + tid * 4];
        acc.x += wc * a.x; acc.y += wc * a.y; acc.z += wc * a.z; acc.w += wc * a.w;
    }
    const float inv = 1.0f / L;
    float4 o = make_float4(acc.x * inv, acc.y * inv, acc.z * inv, acc.w * inv);
    *(float4*)&out[(size_t)b * CHE + tid * 4] = o;
}

extern "C" void kernel_launch(void* const* d_in, const int* in_sizes, int n_in,
                              void* d_out, int out_size, void* d_ws, size_t ws_size,
                              hipStream_t stream)
{
    // d_in: 0=hidden (unused: hid@Wh + b cancels in masked softmax), 1=encoder_outputs,
    //       2=mask, 3=W, 4=b (unused)
    const float* enc  = (const float*)d_in[1];
    const float* mask = (const float*)d_in[2];
    const float* W    = (const float*)d_in[3];
    float* ws  = (float*)d_ws;
    float* out = (float*)d_out;

    hipFuncSetAttribute((const void*)attn_pass1,
                        hipFuncAttributeMaxDynamicSharedMemorySize, (int)SM_BYTES);
    attn_pass1<<<dim3(CNC, CB), CTHR, SM_BYTES, stream>>>(enc, mask, W, ws);
    attn_pass2<<<CB, CTHR, 0, stream>>>(ws, out);
}